// VEConv_45105746542698
// MI455X (gfx1250) — compile-verified
//
#include <hip/hip_runtime.h>
#include <math.h>

typedef __attribute__((ext_vector_type(2))) float v2f;
typedef __attribute__((ext_vector_type(8))) float v8f;

#define WAVES_PER_BLOCK 4   // 128 threads; each wave owns 2 row-tiles (32 edges)

__device__ __forceinline__ v8f wmma4(v2f a, v2f b, v8f c) {
  // V_WMMA_F32_16X16X4_F32: D = A(16x4) * B(4x16) + C(16x16), wave32
  return __builtin_amdgcn_wmma_f32_16x16x4_f32(
      /*neg_a=*/false, a, /*neg_b=*/false, b,
      /*c_mod=*/(short)0, c, /*reuse_a=*/false, /*reuse_b=*/false);
}

// Softplus(beta=0.5, threshold=14): t = 0.5*x; t>14 ? x : 2*log1p(exp(t)).
// Stable fast form: log1p(exp(t)) = max(t,0) + log(1 + exp(-|t|)); for the
// selected branch t<=14 so exp(-|t|) >= 8.3e-7 and log(1+z) via v_log_f32 is
// accurate to ~1e-7 absolute — negligible after the @W2 contraction.
__device__ __forceinline__ float softplus_b05_t14(float u) {
  float t = 0.5f * u;
  float z = __expf(-fabsf(t));
  float sp = 2.0f * (fmaxf(t, 0.0f) + __logf(1.0f + z));
  return (t > 14.0f) ? u : sp;
}

// B fragment: lanes 0-15 need rows {kc,kc+1}, lanes 16-31 rows {kc+2,kc+3} (kc
// already includes +2*hi), column = col.  PACKED layout puts the row pair
// adjacent: Wp[(kc>>1)*128 + 2*col + r] == W[kc + r][col]  -> one b64 load.
template <bool PACKED>
__device__ __forceinline__ v2f loadB(const float* __restrict__ W, int kc, int col) {
  if constexpr (PACKED) {
    return *(const v2f*)(W + (size_t)kc * 64 + (col << 1));
  } else {
    v2f b;
    b.x = W[(size_t)kc * 64 + col];
    b.y = W[(size_t)kc * 64 + 64 + col];
    return b;
  }
}

template <bool PACKED>
__global__ __launch_bounds__(32 * WAVES_PER_BLOCK)
void veconv_wmma_kernel(const float* __restrict__ rbf,
                        const float* __restrict__ edge_f,
                        const float* __restrict__ new_node,
                        const float* __restrict__ W1, const float* __restrict__ b1,
                        const float* __restrict__ W2, const float* __restrict__ b2,
                        const float* __restrict__ W3, const float* __restrict__ b3,
                        const int* __restrict__ src, const int* __restrict__ dst,
                        float* __restrict__ out, int numTiles, int rbfDim) {
  __shared__ float ldsS[WAVES_PER_BLOCK][2][16 * 64];  // 32 KB: softplus staging

  const int lane   = threadIdx.x & 31;
  const int wid    = threadIdx.x >> 5;
  const int laneLo = lane & 15;   // A: M row index; B/C: N column index
  const int hi     = lane >> 4;   // half-wave K-pair selector
  const int tile   = blockIdx.x * WAVES_PER_BLOCK + wid;  // macro-tile (32 edges)
  if (tile >= numTiles) return;   // wave-uniform: EXEC stays all-1s for WMMA
  const int e0 = tile * 32;

  // ---------------- GEMM1: U = rbf[32,RBF] @ W1[RBF,64] ----------------
  v8f acc[2][4];
#pragma unroll
  for (int t = 0; t < 2; ++t)
#pragma unroll
    for (int n = 0; n < 4; ++n) acc[t][n] = (v8f){};

  {
    const float* arow0 = rbf + (size_t)(e0 + laneLo) * rbfDim;
    const float* arow1 = arow0 + (size_t)16 * rbfDim;
    const int ksteps = rbfDim >> 2;
    for (int k = 0; k < ksteps; ++k) {
      const int kc = k * 4 + 2 * hi;
      v2f a0 = *(const v2f*)(arow0 + kc);
      v2f a1 = *(const v2f*)(arow1 + kc);
#pragma unroll
      for (int n = 0; n < 4; ++n) {
        v2f b = loadB<PACKED>(W1, kc, n * 16 + laneLo);
        acc[0][n] = wmma4(a0, b, acc[0][n]);
        acc[1][n] = wmma4(a1, b, acc[1][n]);
      }
    }
  }

  // ---- bias + softplus; stage S[2][16][64] to LDS (C-layout -> row major) ----
#pragma unroll
  for (int t = 0; t < 2; ++t) {
#pragma unroll
    for (int n = 0; n < 4; ++n) {
      const int col = n * 16 + laneLo;
      const float bias = b1[col];
#pragma unroll
      for (int i = 0; i < 8; ++i) {
        ldsS[wid][t][(i + 8 * hi) * 64 + col] = softplus_b05_t14(acc[t][n][i] + bias);
      }
    }
  }

  // ---------------- GEMM2: H = S[32,64] @ W2[64,64] (A from LDS) ----------------
  v8f h[2][4];
#pragma unroll
  for (int t = 0; t < 2; ++t)
#pragma unroll
    for (int n = 0; n < 4; ++n) h[t][n] = (v8f){};

  for (int k = 0; k < 16; ++k) {
    const int kc = k * 4 + 2 * hi;
    v2f a0 = *(const v2f*)(&ldsS[wid][0][laneLo * 64 + kc]);  // ds_load_b64
    v2f a1 = *(const v2f*)(&ldsS[wid][1][laneLo * 64 + kc]);
#pragma unroll
    for (int n = 0; n < 4; ++n) {
      v2f b = loadB<PACKED>(W2, kc, n * 16 + laneLo);
      h[0][n] = wmma4(a0, b, h[0][n]);
      h[1][n] = wmma4(a1, b, h[1][n]);
    }
  }

  // ---------------- GEMM3: E3 = edge_f[32,64] @ W3[64,64] ----------------
#pragma unroll
  for (int t = 0; t < 2; ++t)
#pragma unroll
    for (int n = 0; n < 4; ++n) acc[t][n] = (v8f){};

  {
    const float* erow0 = edge_f + (size_t)(e0 + laneLo) * 64;
    const float* erow1 = erow0 + (size_t)16 * 64;
    for (int k = 0; k < 16; ++k) {
      const int kc = k * 4 + 2 * hi;
      v2f a0 = *(const v2f*)(erow0 + kc);
      v2f a1 = *(const v2f*)(erow1 + kc);
#pragma unroll
      for (int n = 0; n < 4; ++n) {
        v2f b = loadB<PACKED>(W3, kc, n * 16 + laneLo);
        acc[0][n] = wmma4(a0, b, acc[0][n]);
        acc[1][n] = wmma4(a1, b, acc[1][n]);
      }
    }
  }

  // ---- msg = new_node[src]*(H+b2) + (E3+b3); atomic scatter-add to out[dst] ----
#pragma unroll
  for (int t = 0; t < 2; ++t) {
#pragma unroll
    for (int i = 0; i < 8; ++i) {
      const int m = t * 16 + i + 8 * hi;  // edge-in-macro-tile for this C element
      const int sIdx = src[e0 + m];
      const int dIdx = dst[e0 + m];
      const float* nodeRow = new_node + (size_t)sIdx * 64;
      float* outRow = out + (size_t)dIdx * 64;
#pragma unroll
      for (int n = 0; n < 4; ++n) {
        const int col = n * 16 + laneLo;
        const float hv = h[t][n][i] + b2[col];
        const float e3 = acc[t][n][i] + b3[col];
        atomicAdd(outRow + col, nodeRow[col] * hv + e3);
      }
    }
  }
}

// Interleave consecutive K-row pairs of W[K,64] so a B fragment is one b64:
//   Wp[(k>>1)*128 + 2*col + (k&1)] = W[k*64 + col]        (K must be even)
__global__ void repack_pairs_kernel(const float* __restrict__ W,
                                    float* __restrict__ Wp, int total) {
  int t = blockIdx.x * blockDim.x + threadIdx.x;
  if (t >= total) return;
  const int k2  = t >> 7;          // pair index
  const int rem = t & 127;
  const int col = rem >> 1;
  const int r   = rem & 1;
  Wp[t] = W[(size_t)(2 * k2 + r) * 64 + col];
}

// Scalar fallback for E % 32 edges (unused when E is a multiple of 32).
__global__ void veconv_tail_kernel(const float* __restrict__ rbf,
                                   const float* __restrict__ edge_f,
                                   const float* __restrict__ new_node,
                                   const float* __restrict__ W1, const float* __restrict__ b1,
                                   const float* __restrict__ W2, const float* __restrict__ b2,
                                   const float* __restrict__ W3, const float* __restrict__ b3,
                                   const int* __restrict__ src, const int* __restrict__ dst,
                                   float* __restrict__ out, int eStart, int eCount, int rbfDim) {
  int t = blockIdx.x * blockDim.x + threadIdx.x;
  if (t >= eCount * 64) return;
  int e = eStart + t / 64;
  int col = t & 63;
  float h = 0.f;
  for (int j = 0; j < 64; ++j) {
    float u = b1[j];
    for (int k = 0; k < rbfDim; ++k) u += rbf[(size_t)e * rbfDim + k] * W1[k * 64 + j];
    h += softplus_b05_t14(u) * W2[j * 64 + col];
  }
  h += b2[col];
  float e3 = b3[col];
  for (int k = 0; k < 64; ++k) e3 += edge_f[(size_t)e * 64 + k] * W3[k * 64 + col];
  float msg = new_node[(size_t)src[e] * 64 + col] * h + e3;
  atomicAdd(&out[(size_t)dst[e] * 64 + col], msg);
}

extern "C" void kernel_launch(void* const* d_in, const int* in_sizes, int n_in,
                              void* d_out, int out_size, void* d_ws, size_t ws_size,
                              hipStream_t stream) {
  const float* rbf      = (const float*)d_in[0];
  const float* edge_f   = (const float*)d_in[1];
  const float* new_node = (const float*)d_in[2];
  const float* W1       = (const float*)d_in[3];
  const float* b1       = (const float*)d_in[4];
  const float* W2       = (const float*)d_in[5];
  const float* b2       = (const float*)d_in[6];
  const float* W3       = (const float*)d_in[7];
  const float* b3       = (const float*)d_in[8];
  const int*   src      = (const int*)d_in[9];
  const int*   dst      = (const int*)d_in[10];
  float* out = (float*)d_out;

  const int D   = in_sizes[4];        // 64
  const int RBF = in_sizes[3] / D;    // 100
  const int E   = in_sizes[0] / RBF;  // 800000
  (void)D;

  // out is poisoned by the harness; segmented-sum needs zeros.
  hipMemsetAsync(d_out, 0, (size_t)out_size * sizeof(float), stream);

  const int numTiles = E / 32;              // 32-edge macro-tiles per wave
  const int rem      = E & 31;
  const size_t n1 = (size_t)RBF * 64;       // W1 elements
  const size_t n23 = 64 * 64;               // W2 / W3 elements
  const size_t packedFloats = n1 + 2 * n23;
  const bool usePacked =
      (ws_size >= packedFloats * sizeof(float)) && ((RBF & 3) == 0);

  if (numTiles > 0) {
    const int blocks = (numTiles + WAVES_PER_BLOCK - 1) / WAVES_PER_BLOCK;
    if (usePacked) {
      float* W1p = (float*)d_ws;
      float* W2p = W1p + n1;
      float* W3p = W2p + n23;
      repack_pairs_kernel<<<((int)n1 + 255) / 256, 256, 0, stream>>>(W1, W1p, (int)n1);
      repack_pairs_kernel<<<((int)n23 + 255) / 256, 256, 0, stream>>>(W2, W2p, (int)n23);
      repack_pairs_kernel<<<((int)n23 + 255) / 256, 256, 0, stream>>>(W3, W3p, (int)n23);
      veconv_wmma_kernel<true><<<blocks, 32 * WAVES_PER_BLOCK, 0, stream>>>(
          rbf, edge_f, new_node, W1p, b1, W2p, b2, W3p, b3, src, dst, out,
          numTiles, RBF);
    } else {
      veconv_wmma_kernel<false><<<blocks, 32 * WAVES_PER_BLOCK, 0, stream>>>(
          rbf, edge_f, new_node, W1, b1, W2, b2, W3, b3, src, dst, out,
          numTiles, RBF);
    }
  }
  if (rem > 0) {
    const int threads = rem * 64;
    veconv_tail_kernel<<<(threads + 255) / 256, 256, 0, stream>>>(
        rbf, edge_f, new_node, W1, b1, W2, b2, W3, b3, src, dst, out, E - rem, rem, RBF);
  }
}